// GCN_multirelation_36481452212474
// MI455X (gfx1250) — compile-verified
//
#include <hip/hip_runtime.h>

typedef __attribute__((ext_vector_type(2))) float v2f;
typedef __attribute__((ext_vector_type(8))) float v8f;

#define FDIM 256
#define BM 64
#define BN 128
#define BK 64
#define SXP (BK + 4)   // sX row stride: 68 floats -> conflict-free A b64 reads
#define BNP 144        // sW group-row stride in float2 units: 2*144 % 64 == 32
                       // -> lane-halves hit disjoint bank ranges, conflict-free

// ---------------------------------------------------------------------------
// GEMM: H[nrows x 256] = X[nrows x 256] @ W[256 x 256], fp32 WMMA 16x16x4.
// Block = 256 threads = 8 wave32. Block tile 64x128. Wave tile 16x64.
// W staged in LDS pair-interleaved so every B fragment is ONE ds_load_b64.
// ---------------------------------------------------------------------------
__global__ __launch_bounds__(256)
void gemm_f32_wmma(const float* __restrict__ X, const float* __restrict__ W,
                   float* __restrict__ H, int nrows) {
  __shared__ __align__(16) float sX[BM * SXP];        // 64x68  = 17408 B
  __shared__ __align__(16) float sWp[(BK / 2) * BNP * 2]; // 32x144x2 = 36864 B

  const int t       = threadIdx.x;
  const int lane    = t & 31;
  const int wave    = t >> 5;
  const int l16     = lane & 15;
  const int half    = lane >> 4;          // 0: K pair {kk,kk+1}, 1: {kk+2,kk+3}
  const int waveRow = wave & 3;           // 4 row strips of 16
  const int waveCol = wave >> 2;          // 2 col strips of 64
  const int rowBase = blockIdx.x * BM;
  const int colBase = blockIdx.y * BN;
  const int cb      = waveCol * 64 + l16; // this lane's base column in the tile

  v8f acc0 = {}, acc1 = {}, acc2 = {}, acc3 = {};

  for (int kBase = 0; kBase < FDIM; kBase += BK) {
    // --- stage X tile (BM x BK), zero-fill out-of-range rows -------------
#pragma unroll
    for (int it = 0; it < 4; ++it) {
      int idx = t + 256 * it;             // 0..1023 float4 slots
      int row = idx >> 4;
      int c4  = (idx & 15) << 2;
      float4 v = make_float4(0.f, 0.f, 0.f, 0.f);
      int g = rowBase + row;
      if (g < nrows)
        v = *(const float4*)(X + (size_t)g * FDIM + kBase + c4);
      *(float4*)&sX[row * SXP + c4] = v;
    }
    // --- stage W tile (BK x BN), pair-interleaved:
    //     sWp[(g*BNP + n)*2 + (k&1)] = W[kBase + 2g + (k&1)][colBase + n]
#pragma unroll
    for (int it = 0; it < 8; ++it) {
      int idx = t + 256 * it;             // 0..2047 float4 slots
      int k   = idx >> 5;
      int n4  = (idx & 31) << 2;
      float4 v = *(const float4*)(W + (size_t)(kBase + k) * FDIM + colBase + n4);
      float* p = sWp + ((size_t)(k >> 1) * BNP + n4) * 2 + (k & 1);
      p[0] = v.x; p[2] = v.y; p[4] = v.z; p[6] = v.w;
    }
    __syncthreads();

    // --- compute: 16 k-steps of 4, 4 WMMA tiles per wave -----------------
#pragma unroll
    for (int kk = 0; kk < BK; kk += 4) {
      // A fragment: adjacent pair in sX row -> one b64 LDS load
      const int ks = kk + 2 * half;
      v2f a = *(const v2f*)&sX[(waveRow * 16 + l16) * SXP + ks];
      // B fragments: pair-interleaved row -> one b64 LDS load each
      const v2f* wrow = (const v2f*)(sWp + ((kk >> 1) + half) * BNP * 2);
      v2f b0 = wrow[cb];
      v2f b1 = wrow[cb + 16];
      v2f b2 = wrow[cb + 32];
      v2f b3 = wrow[cb + 48];
      acc0 = __builtin_amdgcn_wmma_f32_16x16x4_f32(false, a, false, b0, (short)0, acc0, false, false);
      acc1 = __builtin_amdgcn_wmma_f32_16x16x4_f32(false, a, false, b1, (short)0, acc1, false, false);
      acc2 = __builtin_amdgcn_wmma_f32_16x16x4_f32(false, a, false, b2, (short)0, acc2, false, false);
      acc3 = __builtin_amdgcn_wmma_f32_16x16x4_f32(false, a, false, b3, (short)0, acc3, false, false);
    }
    __syncthreads();
  }

  // --- store: C layout = VGPR v -> M = v + 8*half, N = l16 ---------------
#pragma unroll
  for (int v = 0; v < 8; ++v) {
    int gr = rowBase + waveRow * 16 + v + 8 * half;
    if (gr < nrows) {
      float* p = H + (size_t)gr * FDIM + colBase + waveCol * 64 + l16;
      p[0]  = acc0[v];
      p[16] = acc1[v];
      p[32] = acc2[v];
      p[48] = acc3[v];
    }
  }
}

// ---------------------------------------------------------------------------
// Edge scatter: acc[dst] += h[src] * val.  64 threads per edge, float4 gather,
// 4 hardware fp32 atomics per thread (no-return -> STOREcnt path).
// ---------------------------------------------------------------------------
__global__ __launch_bounds__(256)
void scatter_edges(const float* __restrict__ h, const int* __restrict__ src,
                   const int* __restrict__ dst, const float* __restrict__ val,
                   float* __restrict__ acc, int E) {
  long long idx = (long long)blockIdx.x * blockDim.x + threadIdx.x;
  int e  = (int)(idx >> 6);
  int c4 = (int)(idx & 63) << 2;
  if (e >= E) return;
  int   s = src[e];
  int   d = dst[e];
  float v = val[e];
  float4 hv = *(const float4*)(h + (size_t)s * FDIM + c4);
  float* a = acc + (size_t)d * FDIM + c4;
  unsafeAtomicAdd(a + 0, hv.x * v);
  unsafeAtomicAdd(a + 1, hv.y * v);
  unsafeAtomicAdd(a + 2, hv.z * v);
  unsafeAtomicAdd(a + 3, hv.w * v);
}

__global__ __launch_bounds__(256)
void zero_buf(float* __restrict__ p, long long n4) {
  long long i = (long long)blockIdx.x * blockDim.x + threadIdx.x;
  if (i < n4) *(float4*)(p + 4 * i) = make_float4(0.f, 0.f, 0.f, 0.f);
}

__global__ __launch_bounds__(256)
void bias_relu(const float* __restrict__ acc, const float* __restrict__ b,
               float* __restrict__ o, long long n) {
  long long i = (long long)blockIdx.x * blockDim.x + threadIdx.x;
  if (i < n) o[i] = fmaxf(acc[i] + b[i & (FDIM - 1)], 0.f);
}

__global__ __launch_bounds__(256)
void final_fuse(const float* __restrict__ acc, const float* __restrict__ b,
                const float* __restrict__ x1, float* __restrict__ out, long long n) {
  long long i = (long long)blockIdx.x * blockDim.x + threadIdx.x;
  if (i < n) out[i] = fmaxf(acc[i] + b[i & (FDIM - 1)], 0.f) + x1[i];
}

// ---------------------------------------------------------------------------
extern "C" void kernel_launch(void* const* d_in, const int* in_sizes, int n_in,
                              void* d_out, int out_size, void* d_ws, size_t ws_size,
                              hipStream_t stream) {
  const float* x    = (const float*)d_in[0];
  const int*   esrc = (const int*)d_in[1];
  const int*   edst = (const int*)d_in[2];
  const float* eval = (const float*)d_in[3];
  const float* W1   = (const float*)d_in[4];
  const float* b1   = (const float*)d_in[5];
  const float* W2   = (const float*)d_in[6];
  const float* b2   = (const float*)d_in[7];
  float* out = (float*)d_out;

  const int R  = 4;
  const int Nn = in_sizes[0] / FDIM;
  const int E  = in_sizes[1] / R;

  const size_t nf = (size_t)Nn * FDIM;
  float* h   = (float*)d_ws;      // [N,256] per-relation GEMM output
  float* acc = h + nf;            // [N,256] scatter accumulator
  float* x1  = acc + nf;          // [N,256] layer-1 activation

  dim3 gGrid((Nn + BM - 1) / BM, FDIM / BN);
  const long long n4 = (long long)(nf / 4);
  const int zBlocks = (int)((n4 + 255) / 256);
  const int sBlocks = (int)(((long long)E * 64 + 255) / 256);
  const int pBlocks = (int)(((long long)nf + 255) / 256);

  // ---- layer 1 ----------------------------------------------------------
  zero_buf<<<zBlocks, 256, 0, stream>>>(acc, n4);
  for (int r = 0; r < R; ++r) {
    gemm_f32_wmma<<<gGrid, 256, 0, stream>>>(x, W1 + (size_t)r * FDIM * FDIM, h, Nn);
    scatter_edges<<<sBlocks, 256, 0, stream>>>(h, esrc + (size_t)r * E,
                                               edst + (size_t)r * E,
                                               eval + (size_t)r * E, acc, E);
  }
  bias_relu<<<pBlocks, 256, 0, stream>>>(acc, b1, x1, (long long)nf);

  // ---- layer 2 ----------------------------------------------------------
  zero_buf<<<zBlocks, 256, 0, stream>>>(acc, n4);
  for (int r = 0; r < R; ++r) {
    gemm_f32_wmma<<<gGrid, 256, 0, stream>>>(x1, W2 + (size_t)r * FDIM * FDIM, h, Nn);
    scatter_edges<<<sBlocks, 256, 0, stream>>>(h, esrc + (size_t)r * E,
                                               edst + (size_t)r * E,
                                               eval + (size_t)r * E, acc, E);
  }
  final_fuse<<<pBlocks, 256, 0, stream>>>(acc, b2, x1, out, (long long)nf);
}